// VariableLayer_71614284693528
// MI455X (gfx1250) — compile-verified
//
#include <hip/hip_runtime.h>
#include <stdint.h>

// out[b,n] = input_llr[b,n] + sum_k mask(idx[n,k]) * check[b, idx[n,k]]
// B=256, N=73728, K=8.
//
// One workgroup per batch row b. Stage the full 288KB row check[b,:] into
// LDS with gfx1250 async global->LDS b128 copies (ASYNCcnt, non-temporal:
// each row is read exactly once), then retire all K=8 random gathers per n
// as ds_load_b32 from LDS. Streamed input/output use non-temporal policy so
// the 2.4MB index array (re-read by all 256 blocks) stays L2-resident.
// HBM traffic is the streaming minimum (~229 MB => ~10us @ 23.3 TB/s).

#define TPB 1024  // 32 wave32 waves per workgroup

extern "C" __global__ void __launch_bounds__(TPB)
var_layer_kernel(const float* __restrict__ input_llr,
                 const float* __restrict__ check,
                 const int*   __restrict__ idx,
                 float* __restrict__ out,
                 int N)
{
    extern __shared__ float lds[];   // N floats = 288KB (of 320KB/WGP)
    const int tid = (int)threadIdx.x;
    const int b   = (int)blockIdx.x;
    const float* row = check + (size_t)b * (size_t)N;

    // ---- Stage check[b,:] into LDS via async b128 copies (ASYNCcnt) ----
    // Dynamic LDS begins at groupstaticsize(); raw LDS byte addresses in
    // the asm must match what `lds[]` indexing lowers to.
    const uint32_t ldsBase = __builtin_amdgcn_groupstaticsize();
    const int nvec = N >> 2;                 // 18432 b128 transfers
    for (int i = tid; i < nvec; i += TPB) {  // 18 iterations, no tail
        uint32_t laddr = ldsBase + ((uint32_t)i << 4);
        uint64_t gaddr = (uint64_t)(uintptr_t)(row + ((size_t)i << 2));
        asm volatile("global_load_async_to_lds_b128 %0, %1, off th:TH_LOAD_NT"
                     :: "v"(laddr), "v"(gaddr)
                     : "memory");
    }

    // Overlap: pull this thread's first index tiles toward the WGP while
    // the async DMA drains (global_prefetch_b8).
    __builtin_prefetch(idx + (size_t)tid * 8, 0, 0);
    __builtin_prefetch(idx + (size_t)(tid + TPB) * 8, 0, 0);

    asm volatile("s_wait_asynccnt 0x0" ::: "memory");
    __syncthreads();

    // ---- Gather-reduce: 8 LDS gathers + coalesced stream in/out ----
    // Branch-free masked gather: clamp address, cndmask the value.
#define GATHER(vexp, acc) { int v_ = (vexp); int safe_ = (v_ < 0) ? 0 : v_; \
                            float g_ = lds[safe_]; acc += (v_ >= 0) ? g_ : 0.0f; }

    const size_t rowOff = (size_t)b * (size_t)N;
    int n = tid;
    // 2x unrolled main body: two independent 8-gather chains in flight.
    for (; n + TPB < N; n += 2 * TPB) {
        const int4* ipA = (const int4*)(idx + ((size_t)n << 3));        // 32B aligned
        const int4* ipB = (const int4*)(idx + ((size_t)(n + TPB) << 3));
        const int4 a0 = ipA[0], a1 = ipA[1];
        const int4 b0 = ipB[0], b1 = ipB[1];
        float sA = 0.0f, sB = 0.0f;
        GATHER(a0.x, sA) GATHER(a0.y, sA) GATHER(a0.z, sA) GATHER(a0.w, sA)
        GATHER(b0.x, sB) GATHER(b0.y, sB) GATHER(b0.z, sB) GATHER(b0.w, sB)
        GATHER(a1.x, sA) GATHER(a1.y, sA) GATHER(a1.z, sA) GATHER(a1.w, sA)
        GATHER(b1.x, sB) GATHER(b1.y, sB) GATHER(b1.z, sB) GATHER(b1.w, sB)
        const float inA = __builtin_nontemporal_load(input_llr + rowOff + n);
        const float inB = __builtin_nontemporal_load(input_llr + rowOff + n + TPB);
        __builtin_nontemporal_store(inA + sA, out + rowOff + n);
        __builtin_nontemporal_store(inB + sB, out + rowOff + n + TPB);
    }
    // Tail (runs when N/TPB is odd).
    for (; n < N; n += TPB) {
        const int4* ip = (const int4*)(idx + ((size_t)n << 3));
        const int4 i0 = ip[0], i1 = ip[1];
        float s = 0.0f;
        GATHER(i0.x, s) GATHER(i0.y, s) GATHER(i0.z, s) GATHER(i0.w, s)
        GATHER(i1.x, s) GATHER(i1.y, s) GATHER(i1.z, s) GATHER(i1.w, s)
        const float in = __builtin_nontemporal_load(input_llr + rowOff + n);
        __builtin_nontemporal_store(in + s, out + rowOff + n);
    }
#undef GATHER
}

extern "C" void kernel_launch(void* const* d_in, const int* in_sizes, int n_in,
                              void* d_out, int out_size, void* d_ws, size_t ws_size,
                              hipStream_t stream) {
    (void)n_in; (void)out_size; (void)d_ws; (void)ws_size;
    const int K = 8;
    const int N = in_sizes[2] / K;          // 73728
    const int B = in_sizes[0] / N;          // 256

    const float* input_llr = (const float*)d_in[0];
    const float* check     = (const float*)d_in[1];
    const int*   idxp      = (const int*)d_in[2];
    float* out = (float*)d_out;

    const size_t shmem = (size_t)N * sizeof(float);  // 294912 B
    // Allow >64KB dynamic LDS (CDNA5 WGP has 320KB). Not a stream op, so
    // it is safe under graph capture; errors intentionally ignored.
    (void)hipFuncSetAttribute((const void*)var_layer_kernel,
                              hipFuncAttributeMaxDynamicSharedMemorySize,
                              (int)shmem);

    hipLaunchKernelGGL(var_layer_kernel, dim3(B), dim3(TPB), shmem, stream,
                       input_llr, check, idxp, out, N);
}